// SimpleGNNRisk_62191126446559
// MI455X (gfx1250) — compile-verified
//
#include <hip/hip_runtime.h>
#include <hip/hip_bf16.h>

// GCN risk model for MI455X (gfx1250, wave32).
// Fixed problem dims from the reference: C=128 (input feats), H=64 (hidden).
#define FEAT_C 128
#define FEAT_H 64

typedef __attribute__((ext_vector_type(2))) float v2f;
typedef __attribute__((ext_vector_type(8))) float v8f;

// ---------------------------------------------------------------------------
// Degree / normalization
// ---------------------------------------------------------------------------
__global__ void deg_init_kernel(float* dinv, int n) {
    int i = blockIdx.x * blockDim.x + threadIdx.x;
    if (i < n) dinv[i] = 1.0f;  // self-loop contributes 1 to every node degree
}

__global__ void deg_edges_kernel(const int* __restrict__ dst, float* dinv, int e) {
    int i = blockIdx.x * blockDim.x + threadIdx.x;
    if (i < e) unsafeAtomicAdd(&dinv[dst[i]], 1.0f);  // global_atomic_add_f32
}

__global__ void deg_finalize_kernel(float* dinv, int n) {
    int i = blockIdx.x * blockDim.x + threadIdx.x;
    if (i < n) dinv[i] = rsqrtf(dinv[i]);  // deg >= 1 always (self loops)
}

// ---------------------------------------------------------------------------
// Dense GEMM via V_WMMA_F32_16X16X4_F32 (fp32 WMMA, full precision).
// out[rows,64] = X[rows,K] @ W[K,64].  One CTA = 16 rows, 4 waves, each wave
// owns one 16-wide column tile.  rows must be a multiple of 16 (100000 = 6250*16).
//
// A 16x4 f32 layout (ISA 7.12.2): lanes 0-15 hold M=lane, VGPR0=K0,VGPR1=K1;
//                                 lanes 16-31 hold M=lane-16, VGPR0=K2,VGPR1=K3.
// B 4x16 f32 layout (mirrored):   lanes 0-15 hold N=lane, K0/K1;
//                                 lanes 16-31 hold N=lane-16, K2/K3.
// D 16x16 f32 layout: VGPR r -> M=r (lanes 0-15), M=r+8 (lanes 16-31), N=lane%16.
// ---------------------------------------------------------------------------
template <int K>
__global__ void gemm_wmma_f32_kernel(const float* __restrict__ X,
                                     const float* __restrict__ W,
                                     float* __restrict__ out) {
    const int lane = threadIdx.x & 31;
    const int wave = threadIdx.x >> 5;        // 0..3 -> column tile
    const int row0 = blockIdx.x * 16;
    const int col0 = wave * 16;
    const int m    = lane & 15;
    const int koff = (lane >> 4) * 2;         // 0 for lanes 0-15, 2 for 16-31

    const float* __restrict__ xrow = X + (size_t)(row0 + m) * K;

    v8f c = {};
#pragma unroll
    for (int k = 0; k < K; k += 4) {
        v2f a, b;
        a.x = xrow[k + koff];                       // contiguous pair (b64 load)
        a.y = xrow[k + koff + 1];
        b.x = W[(size_t)(k + koff) * FEAT_H + col0 + m];
        b.y = W[(size_t)(k + koff + 1) * FEAT_H + col0 + m];
        c = __builtin_amdgcn_wmma_f32_16x16x4_f32(
                /*neg_a=*/false, a, /*neg_b=*/false, b,
                /*c_mod=*/(short)0, c, /*reuse_a=*/false, /*reuse_b=*/false);
    }

    const int mbase = (lane < 16) ? 0 : 8;
#pragma unroll
    for (int r = 0; r < 8; ++r) {
        out[(size_t)(row0 + mbase + r) * FEAT_H + col0 + m] = c[r];
    }
}

// ---------------------------------------------------------------------------
// Zero a float buffer (agg targets) — avoids relying on poisoned workspace.
// ---------------------------------------------------------------------------
__global__ void zero_kernel(float* p, long long count) {
    long long i = (long long)blockIdx.x * blockDim.x + threadIdx.x;
    if (i < count) p[i] = 0.0f;
}

// ---------------------------------------------------------------------------
// Edge scatter: one wave per edge. Each lane moves 2 of the 64 features.
// agg[dst] += h[src] * dinv[src]*dinv[dst]  via fp32 HW atomics (lands in L2;
// both h and agg fit in the 192MB L2, so this stays on-chip).
// ---------------------------------------------------------------------------
__global__ void scatter_edges_kernel(const float* __restrict__ h,
                                     const int* __restrict__ src,
                                     const int* __restrict__ dst,
                                     const float* __restrict__ dinv,
                                     float* agg, int e) {
    int wid  = (int)(((long long)blockIdx.x * blockDim.x + threadIdx.x) >> 5);
    int lane = threadIdx.x & 31;
    if (wid >= e) return;
    int s = src[wid];
    int d = dst[wid];
    float nrm = dinv[s] * dinv[d];
    const float2 v = reinterpret_cast<const float2*>(h + (size_t)s * FEAT_H)[lane];
    float* base = agg + (size_t)d * FEAT_H + lane * 2;
    unsafeAtomicAdd(base + 0, v.x * nrm);
    unsafeAtomicAdd(base + 1, v.y * nrm);
}

// ---------------------------------------------------------------------------
// Per-node finalize: add self-loop term (norm = dinv[n]^2), bias, ReLU.
// In-place on agg.
// ---------------------------------------------------------------------------
__global__ void finalize_node_kernel(float* agg, const float* __restrict__ h,
                                     const float* __restrict__ dinv,
                                     const float* __restrict__ bias, int n) {
    long long idx = (long long)blockIdx.x * blockDim.x + threadIdx.x;
    if (idx >= (long long)n * FEAT_H) return;
    int node = (int)(idx >> 6);
    int f    = (int)(idx & 63);
    float di = dinv[node];
    float v  = agg[idx] + h[idx] * di * di + bias[f];
    agg[idx] = fmaxf(v, 0.0f);
}

// ---------------------------------------------------------------------------
// MLP head: one wave per node.  Lane j owns hidden unit j (H/2 = 32 = wave32).
// hj = relu(dot(row, Wm1[:,j]) + bm1[j]);  out = sigmoid(sum_j hj*Wm2[j] + bm2)
// ---------------------------------------------------------------------------
__global__ void mlp_head_kernel(const float* __restrict__ h,
                                const float* __restrict__ Wm1,
                                const float* __restrict__ bm1,
                                const float* __restrict__ Wm2,
                                const float* __restrict__ bm2,
                                float* __restrict__ out, int n) {
    int node = (int)(((long long)blockIdx.x * blockDim.x + threadIdx.x) >> 5);
    int lane = threadIdx.x & 31;
    if (node >= n) return;
    const float* row = h + (size_t)node * FEAT_H;
    float acc = bm1[lane];
#pragma unroll
    for (int k = 0; k < FEAT_H; ++k) {
        acc = fmaf(row[k], Wm1[k * (FEAT_H / 2) + lane], acc);  // coalesced Wm1
    }
    acc = fmaxf(acc, 0.0f) * Wm2[lane];
#pragma unroll
    for (int off = 16; off > 0; off >>= 1) {
        acc += __shfl_down(acc, off, 32);  // wave32 reduction
    }
    if (lane == 0) {
        float z = acc + bm2[0];
        out[node] = 1.0f / (1.0f + __expf(-z));
    }
}

// ---------------------------------------------------------------------------
// Launch
// ---------------------------------------------------------------------------
extern "C" void kernel_launch(void* const* d_in, const int* in_sizes, int n_in,
                              void* d_out, int out_size, void* d_ws, size_t ws_size,
                              hipStream_t stream) {
    const float* x   = (const float*)d_in[0];   // [N, 128]
    const int*   ei  = (const int*)d_in[1];     // [2, E] row-major: src then dst
    const float* W1  = (const float*)d_in[2];   // [128, 64]
    const float* b1  = (const float*)d_in[3];   // [64]
    const float* W2  = (const float*)d_in[4];   // [64, 64]
    const float* b2  = (const float*)d_in[5];   // [64]
    const float* Wm1 = (const float*)d_in[6];   // [64, 32]
    const float* bm1 = (const float*)d_in[7];   // [32]
    const float* Wm2 = (const float*)d_in[8];   // [32, 1]
    const float* bm2 = (const float*)d_in[9];   // [1]

    const int N = out_size;             // 100000
    const int E = in_sizes[1] / 2;      // 1600000
    const int* src = ei;
    const int* dst = ei + E;

    // Workspace layout (all fp32): dinv[N] | bufA[N*64] | bufB[N*64]
    float* dinv = (float*)d_ws;
    float* bufA = dinv + N;                       // N*4 = 400000 B, 16B-aligned
    float* bufB = bufA + (size_t)N * FEAT_H;
    (void)ws_size; (void)n_in;

    const long long nodeFeat = (long long)N * FEAT_H;
    const int TB = 256;
    const int gridN   = (N + TB - 1) / TB;
    const int gridE   = (E + TB - 1) / TB;
    const int gridNF  = (int)((nodeFeat + TB - 1) / TB);
    const int gridEW  = (int)(((long long)E * 32 + TB - 1) / TB);  // 1 wave/edge
    const int gridNW  = (int)(((long long)N * 32 + TB - 1) / TB);  // 1 wave/node
    const int gridGemm = N / 16;                                   // 6250

    // 1) symmetric GCN normalization: dinv = rsqrt(deg + selfloop)
    deg_init_kernel<<<gridN, TB, 0, stream>>>(dinv, N);
    deg_edges_kernel<<<gridE, TB, 0, stream>>>(dst, dinv, E);
    deg_finalize_kernel<<<gridN, TB, 0, stream>>>(dinv, N);

    // 2) layer 1: bufA = x @ W1 (WMMA f32), scatter -> bufB, relu(+b1)
    gemm_wmma_f32_kernel<FEAT_C><<<gridGemm, 128, 0, stream>>>(x, W1, bufA);
    zero_kernel<<<gridNF, TB, 0, stream>>>(bufB, nodeFeat);
    scatter_edges_kernel<<<gridEW, TB, 0, stream>>>(bufA, src, dst, dinv, bufB, E);
    finalize_node_kernel<<<gridNF, TB, 0, stream>>>(bufB, bufA, dinv, b1, N);

    // 3) layer 2: bufA = bufB @ W2, scatter -> bufB (re-zeroed), relu(+b2)
    gemm_wmma_f32_kernel<FEAT_H><<<gridGemm, 128, 0, stream>>>(bufB, W2, bufA);
    zero_kernel<<<gridNF, TB, 0, stream>>>(bufB, nodeFeat);
    scatter_edges_kernel<<<gridEW, TB, 0, stream>>>(bufA, src, dst, dinv, bufB, E);
    finalize_node_kernel<<<gridNF, TB, 0, stream>>>(bufB, bufA, dinv, b2, N);

    // 4) MLP head -> sigmoid -> d_out[N]
    mlp_head_kernel<<<gridNW, TB, 0, stream>>>(bufB, Wm1, bm1, Wm2, bm2,
                                               (float*)d_out, N);
}